// Metric_56985626083917
// MI455X (gfx1250) — compile-verified
//
#include <hip/hip_runtime.h>
#include <hip/hip_bf16.h>

typedef float v2f __attribute__((ext_vector_type(2)));
typedef float v8f __attribute__((ext_vector_type(8)));

#define B_BATCH 8
#define NPTS    4096
#define KSEL    2048          // int(0.5 * 4096)
#define WEIGHT  3.0f
#define RT      4             // row tiles (of 16) per block -> 64-row strip
#define CT      32            // column tiles (of 16) per wave
#define INF_I   0x7F800000    // +inf bits; > any finite float's bits (int order)

__device__ __forceinline__ int imin2(int a, int b) { return a < b ? a : b; }  // v_min_i32
__device__ __forceinline__ unsigned clamp0(int bits) {                        // v_max_i32
    return (unsigned)(bits > 0 ? bits : 0);
}

// ---------------------------------------------------------------------------
// Kernel 0: init workspace mins to +inf bits, zero the output scalar.
// ---------------------------------------------------------------------------
__global__ void chamfer_init(unsigned int* __restrict__ mins, float* __restrict__ out) {
    int idx = blockIdx.x * blockDim.x + threadIdx.x;
    if (idx < 2 * B_BATCH * NPTS) mins[idx] = (unsigned)INF_I;
    if (idx == 0) out[0] = 0.0f;
}

// ---------------------------------------------------------------------------
// Kernel 1: pairwise sq-dist tiles via V_WMMA_F32_16X16X4_F32.
//   A[M,:] = [-2px, -2py, -2pz, |p|^2],  B[:,N] = [gx, gy, gz, 1],
//   C preloaded with |g|^2  =>  D[M,N] = |p-g|^2 directly (maybe ~-eps).
// Min accumulation on RAW signed bit patterns (v_min_i32): equals
// float-min-then-clamp once final max_i32(.,0) is applied at the atomic.
// Grid: (row_strip=64, batch=8), block 256 threads = 8 waves.
// ---------------------------------------------------------------------------
__global__ __launch_bounds__(256)
void chamfer_tiles(const float* __restrict__ pred, const float* __restrict__ gt,
                   unsigned int* __restrict__ min1,   // [B][N] pred->gt
                   unsigned int* __restrict__ min2) { // [B][M] gt->pred
    const int lane = threadIdx.x & 31;
    const int wave = threadIdx.x >> 5;
    const int b    = blockIdx.y;
    const int rs   = blockIdx.x;                 // 64-row strip index
    const bool hi  = lane >= 16;
    const int l16  = lane & 15;

    const size_t pbase = (size_t)b * NPTS * 3;

    // --- 4 A-matrices: 16 pred rows each (both halves read same 16 rows) ---
    v2f a[RT];
    #pragma unroll
    for (int r = 0; r < RT; ++r) {
        const int row = rs * (RT * 16) + r * 16 + l16;
        const float* pp = pred + pbase + (size_t)row * 3;
        const float px = pp[0], py = pp[1], pz = pp[2];
        const float pn = px * px + py * py + pz * pz;
        a[r].x = hi ? (-2.0f * pz) : (-2.0f * px);   // K=2 (hi) / K=0 (lo)
        a[r].y = hi ? pn           : (-2.0f * py);   // K=3 (hi) / K=1 (lo)
    }

    int rmin[RT][8];
    #pragma unroll
    for (int r = 0; r < RT; ++r)
        #pragma unroll
        for (int v = 0; v < 8; ++v) rmin[r][v] = INF_I;

    const int ct0 = wave * CT;

    // fixed per-lane base; per-iteration offset is wave-uniform (SALU-friendly)
    const char* gwave = (const char*)(gt + pbase + (size_t)(ct0 * 16 + l16) * 3);

    // software-pipelined gt point load (wraps on last iteration; harmless)
    float gx, gy, gz;
    { const float* g0 = (const float*)gwave; gx = g0[0]; gy = g0[1]; gz = g0[2]; }

    for (int i = 0; i < CT; ++i) {
        // prefetch next tile's point: uniform byte offset ((i+1)&31)*192
        const float* gpn = (const float*)(gwave + ((i + 1) & (CT - 1)) * (16 * 3 * 4));
        const float nx = gpn[0], ny = gpn[1], nz = gpn[2];

        const float gnorm = gx * gx + gy * gy + gz * gz;
        v2f bm;
        bm.x = hi ? gz   : gx;                       // B row K=2 (hi) / K=0 (lo)
        bm.y = hi ? 1.0f : gy;                       // B row K=3 (hi) / K=1 (lo)

        v8f c0;                                      // C[v][lane] = |g(N=lane%16)|^2
        #pragma unroll
        for (int v = 0; v < 8; ++v) c0[v] = gnorm;

        int colm = INF_I;                            // partial col min (this half)
        #pragma unroll
        for (int r = 0; r < RT; ++r) {
            const v8f d = __builtin_amdgcn_wmma_f32_16x16x4_f32(
                    false, a[r], false, bm, (short)0, c0, false, false);
            #pragma unroll
            for (int v = 0; v < 8; ++v) {
                const int s = __float_as_int(d[v]);  // raw bits, no clamp
                rmin[r][v] = imin2(rmin[r][v], s);   // v_min_i32
                colm       = imin2(colm, s);         // v_min(3)_i32
            }
        }

        // clamp once; both halves carry partials -> atomicMin combines them
        const int col = (ct0 + i) * 16 + l16;
        atomicMin(&min2[b * NPTS + col], clamp0(colm));

        gx = nx; gy = ny; gz = nz;
    }

    // --- row mins: reduce across the 16 lanes of each half (off hot path) ---
    #pragma unroll
    for (int r = 0; r < RT; ++r) {
        #pragma unroll
        for (int v = 0; v < 8; ++v) {
            int rv = rmin[r][v];
            rv = imin2(rv, __shfl_xor(rv, 8, 16));
            rv = imin2(rv, __shfl_xor(rv, 4, 16));
            rv = imin2(rv, __shfl_xor(rv, 2, 16));
            rv = imin2(rv, __shfl_xor(rv, 1, 16));
            if (l16 == 0) {
                const int m = rs * (RT * 16) + r * 16 + v + (hi ? 8 : 0);
                atomicMin(&min1[b * NPTS + m], clamp0(rv));
            }
        }
    }
}

// ---------------------------------------------------------------------------
// Kernel 2: per (batch, direction): dist = sqrt(min), bitonic sort 4096 in
// LDS, sum all + top-2048, accumulate (mean + 3*topmean)/8 into out[0].
// Grid: 16 blocks x 1024 threads.
// ---------------------------------------------------------------------------
__global__ __launch_bounds__(1024)
void chamfer_finalize(const unsigned int* __restrict__ min1,
                      const unsigned int* __restrict__ min2,
                      float* __restrict__ out) {
    __shared__ float s[NPTS];
    __shared__ float red[1024];
    __shared__ float sums[2];

    const int tid = threadIdx.x;
    const int dir = blockIdx.x & 1;
    const int b   = blockIdx.x >> 1;
    const unsigned int* src = (dir == 0 ? min1 : min2) + b * NPTS;

    for (int i = tid; i < NPTS; i += 1024)
        s[i] = sqrtf(__uint_as_float(src[i]));      // bits already clamped >= 0
    __syncthreads();

    // bitonic sort, ascending (top-k = tail)
    for (int k = 2; k <= NPTS; k <<= 1) {
        for (int j = k >> 1; j > 0; j >>= 1) {
            for (int i = tid; i < NPTS; i += 1024) {
                const int ixj = i ^ j;
                if (ixj > i) {
                    const float av = s[i], bv = s[ixj];
                    const bool asc = (i & k) == 0;
                    if (asc ? (av > bv) : (av < bv)) { s[i] = bv; s[ixj] = av; }
                }
            }
            __syncthreads();
        }
    }

    float sAll = 0.0f, sTop = 0.0f;
    for (int i = tid; i < NPTS; i += 1024) {
        const float v = s[i];
        sAll += v;
        if (i >= NPTS - KSEL) sTop += v;
    }

    red[tid] = sAll; __syncthreads();
    for (int st = 512; st > 0; st >>= 1) {
        if (tid < st) red[tid] += red[tid + st];
        __syncthreads();
    }
    if (tid == 0) sums[0] = red[0];
    __syncthreads();

    red[tid] = sTop; __syncthreads();
    for (int st = 512; st > 0; st >>= 1) {
        if (tid < st) red[tid] += red[tid + st];
        __syncthreads();
    }
    if (tid == 0) sums[1] = red[0];
    __syncthreads();

    if (tid == 0) {
        const float meanD = sums[0] / (float)NPTS;
        const float meanT = sums[1] / (float)KSEL;
        atomicAdd(out, (meanD + WEIGHT * meanT) / (float)B_BATCH);
    }
}

// ---------------------------------------------------------------------------
extern "C" void kernel_launch(void* const* d_in, const int* in_sizes, int n_in,
                              void* d_out, int out_size, void* d_ws, size_t ws_size,
                              hipStream_t stream) {
    const float* pred = (const float*)d_in[0];
    const float* gt   = (const float*)d_in[1];
    float* out        = (float*)d_out;

    unsigned int* min1 = (unsigned int*)d_ws;           // [8][4096]
    unsigned int* min2 = min1 + B_BATCH * NPTS;         // [8][4096]

    chamfer_init<<<(2 * B_BATCH * NPTS + 255) / 256, 256, 0, stream>>>(min1, out);
    chamfer_tiles<<<dim3(NPTS / (RT * 16), B_BATCH), 256, 0, stream>>>(pred, gt, min1, min2);
    chamfer_finalize<<<2 * B_BATCH, 1024, 0, stream>>>(min1, min2, out);
}